// Attention_40166534152481
// MI455X (gfx1250) — compile-verified
//
#include <hip/hip_runtime.h>
#include <math.h>

// ---------------------------------------------------------------------------
// Problem constants (from reference): B=4, M=64, K=32, C=256
// ch = K*K = 1024, red = 2 -> H = 512
// d_in[0] = x  (B*M*K*C f32)
// d_in[1] = W1 (H*CH f32)
// d_in[2] = W2 (CH*H f32)
// d_out   = out (B*M*K*C f32) ++ adj (B*M*K*K f32)
// ---------------------------------------------------------------------------
#define B_  4
#define M_  64
#define K_  32
#define C_  256
#define CH_ 1024   // K*K
#define H_  512    // CH/2

typedef __attribute__((ext_vector_type(2))) float v2f;
typedef __attribute__((ext_vector_type(8))) float v8f;

// FP32 WMMA: D(16x16,f32) = A(16x4,f32) * B(4x16,f32) + C
// 8-arg form: (neg_a, A, neg_b, B, c_mod, C, reuse_a, reuse_b)
static __device__ __forceinline__ v8f wmma_f32_k4(v2f a, v2f b, v8f c) {
  return __builtin_amdgcn_wmma_f32_16x16x4_f32(
      false, a, false, b, (short)0, c, false, false);
}

// ---------------------------------------------------------------------------
// Kernel 1: per-(b,m) tile.  cov/adj via WMMA, pooled features -> scratch p.
// grid = B*M = 256 blocks, 128 threads (4 wave32).
// ---------------------------------------------------------------------------
__global__ void __launch_bounds__(128) tile_stats_kernel(
    const float* __restrict__ x, float* __restrict__ adj_out,
    float* __restrict__ p) {
  __shared__ float xs[K_][C_];        // 32 KB, original x tile
  __shared__ float rowsum[K_];
  __shared__ float covs[K_][K_ + 1];  // padded vs bank conflicts
  __shared__ float stds[K_];

  const int t   = blockIdx.x;         // b*M + m
  const int b   = t / M_;
  const int m   = t % M_;
  const int tid = threadIdx.x;
  const float* xp = x + (size_t)t * K_ * C_;

  for (int idx = tid; idx < K_ * C_; idx += 128)
    xs[idx >> 8][idx & (C_ - 1)] = xp[idx];
  __syncthreads();

  if (tid < K_) {
    float s = 0.0f;
    for (int c = 0; c < C_; ++c) s += xs[tid][c];
    rowsum[tid] = s;
  }
  __syncthreads();

  // --- covariance: cen @ cen^T via v_wmma_f32_16x16x4_f32, K = 256 ---
  const int wave = tid >> 5, lane = tid & 31;
  const int half = lane >> 4, l = lane & 15;
  const int wi = (wave >> 1) * 16;    // output row tile
  const int wj = (wave & 1) * 16;     // output col tile
  const float invC = 1.0f / (float)C_;
  {
    const float mua = rowsum[wi + l] * invC;  // cen computed inline
    const float mub = rowsum[wj + l] * invC;
    v8f acc = {};
    for (int k0 = 0; k0 < C_; k0 += 4) {
      const int ka = k0 + 2 * half;   // half-wave carries K+2,K+3
      v2f av, bv;
      av.x = xs[wi + l][ka]     - mua;
      av.y = xs[wi + l][ka + 1] - mua;
      bv.x = xs[wj + l][ka]     - mub;   // B[k][n] = cen[n][k]
      bv.y = xs[wj + l][ka + 1] - mub;
      acc = wmma_f32_k4(av, bv, acc);
    }
    const float sc = 1.0f / (float)(C_ - 1);
#pragma unroll
    for (int v = 0; v < 8; ++v)       // D: row = v + 8*half, col = l
      covs[wi + v + 8 * half][wj + l] = acc[v] * sc;
  }
  __syncthreads();

  if (tid < K_) stds[tid] = sqrtf(covs[tid][tid]);  // sum(cen^2)/(C-1)
  __syncthreads();

  for (int idx = tid; idx < K_ * K_; idx += 128) {
    const int i = idx >> 5, j = idx & 31;
    adj_out[(size_t)t * (K_ * K_) + idx] = covs[i][j] / (stds[i] * stds[j]);
  }

  // --- pooled[i][j] = 0.5*(mean_i + mean_c(pos(i,j,c)*x[j][c])) ---
  // pos = 1 iff x[i][c]==x[j][c] else 2 (exact fp compare, as reference)
  for (int pair = tid; pair < K_ * K_; pair += 128) {
    const int i = pair >> 5, j = pair & 31;
    float acc2 = 0.0f;
    for (int c = 0; c < C_; ++c) {
      const float xj = xs[j][c];
      acc2 += (xs[i][c] == xj) ? xj : 2.0f * xj;
    }
    const float pooled = 0.5f * (rowsum[i] * invC + acc2 * invC);
    // p[b][i*K + j][m]
    p[((size_t)b * CH_ + pair) * M_ + m] = pooled;
  }
}

// ---------------------------------------------------------------------------
// Kernel 2: h = gelu(W1 @ p).  One 16x16 output tile per wave, K = 1024.
// tiles = B * (H/16) * (M/16) = 4*32*4 = 512 -> 64 blocks x 8 waves.
// ---------------------------------------------------------------------------
__global__ void __launch_bounds__(256) mlp1_kernel(
    const float* __restrict__ W1, const float* __restrict__ p,
    float* __restrict__ hbuf) {
  const int wave = threadIdx.x >> 5, lane = threadIdx.x & 31;
  const int half = lane >> 4, l = lane & 15;
  const int tile = blockIdx.x * 8 + wave;       // 0..511
  const int b    = tile >> 7;                   // tile / 128
  const int rem  = tile & 127;
  const int o0   = (rem >> 2) * 16;
  const int m0   = (rem & 3) * 16;

  const float* w  = W1 + (size_t)(o0 + l) * CH_;
  const float* pp = p + (size_t)b * CH_ * M_ + (m0 + l);

  v8f acc = {};
  for (int k0 = 0; k0 < CH_; k0 += 4) {
    const int ka = k0 + 2 * half;
    v2f av, bv;
    av.x = w[ka];
    av.y = w[ka + 1];
    bv.x = pp[(size_t)ka * M_];        // p rows are m-contiguous -> coalesced
    bv.y = pp[(size_t)(ka + 1) * M_];
    acc = wmma_f32_k4(av, bv, acc);
  }
#pragma unroll
  for (int v = 0; v < 8; ++v) {
    const float val = acc[v];
    const float g = 0.5f * val * (1.0f + erff(val * 0.70710678118654752f));
    hbuf[((size_t)b * H_ + (o0 + v + 8 * half)) * M_ + (m0 + l)] = g;
  }
}

// ---------------------------------------------------------------------------
// Kernel 3: e = sigmoid(W2 @ h).  K = 512.
// tiles = B * (CH/16) * (M/16) = 4*64*4 = 1024 -> 128 blocks x 8 waves.
// ---------------------------------------------------------------------------
__global__ void __launch_bounds__(256) mlp2_kernel(
    const float* __restrict__ W2, const float* __restrict__ hbuf,
    float* __restrict__ e) {
  const int wave = threadIdx.x >> 5, lane = threadIdx.x & 31;
  const int half = lane >> 4, l = lane & 15;
  const int tile = blockIdx.x * 8 + wave;       // 0..1023
  const int b    = tile >> 8;                   // tile / 256
  const int rem  = tile & 255;
  const int o0   = (rem >> 2) * 16;
  const int m0   = (rem & 3) * 16;

  const float* w  = W2 + (size_t)(o0 + l) * H_;
  const float* hp = hbuf + (size_t)b * H_ * M_ + (m0 + l);

  v8f acc = {};
  for (int k0 = 0; k0 < H_; k0 += 4) {
    const int ka = k0 + 2 * half;
    v2f av, bv;
    av.x = w[ka];
    av.y = w[ka + 1];
    bv.x = hp[(size_t)ka * M_];
    bv.y = hp[(size_t)(ka + 1) * M_];
    acc = wmma_f32_k4(av, bv, acc);
  }
#pragma unroll
  for (int v = 0; v < 8; ++v) {
    const float val = acc[v];
    const float s = 1.0f / (1.0f + expf(-val));
    e[((size_t)b * CH_ + (o0 + v + 8 * half)) * M_ + (m0 + l)] = s;
  }
}

// ---------------------------------------------------------------------------
// Kernel 4: masked softmax + out = attention @ x via WMMA (K = 32).
// grid = 256 blocks, 128 threads (4 waves); each wave: 8 output 16x16 tiles.
// ---------------------------------------------------------------------------
__global__ void __launch_bounds__(128) attn_out_kernel(
    const float* __restrict__ x, const float* __restrict__ adj_in,
    const float* __restrict__ e, float* __restrict__ out) {
  __shared__ float xs[K_][C_];
  __shared__ float att[K_][K_ + 1];

  const int t   = blockIdx.x;
  const int b   = t / M_;
  const int m   = t % M_;
  const int tid = threadIdx.x;
  const float* xp = x + (size_t)t * K_ * C_;

  for (int idx = tid; idx < K_ * C_; idx += 128)
    xs[idx >> 8][idx & (C_ - 1)] = xp[idx];
  for (int idx = tid; idx < K_ * K_; idx += 128) {
    const int i = idx >> 5, j = idx & 31;
    const float a  = adj_in[(size_t)t * (K_ * K_) + idx];
    const float ev = e[((size_t)b * CH_ + idx) * M_ + m];
    att[i][j] = (a > 0.0f) ? ev : -9.0e15f;
  }
  __syncthreads();

  if (tid < K_) {                      // row-wise masked softmax
    float mx = -3.0e38f;
    for (int j = 0; j < K_; ++j) mx = fmaxf(mx, att[tid][j]);
    float s = 0.0f;
    for (int j = 0; j < K_; ++j) {
      const float ev = expf(att[tid][j] - mx);
      att[tid][j] = ev;
      s += ev;
    }
    const float inv = 1.0f / s;
    for (int j = 0; j < K_; ++j) att[tid][j] *= inv;
  }
  __syncthreads();

  // out(32x256) = att(32x32) @ x(32x256), K = 32 -> 8 WMMAs per tile
  const int wave = tid >> 5, lane = tid & 31;
  const int half = lane >> 4, l = lane & 15;
  const int i0 = (wave >> 1) * 16;
  float* op = out + (size_t)t * K_ * C_;

  for (int tt = 0; tt < 8; ++tt) {
    const int c0 = (wave & 1) * 16 + tt * 32;
    v8f acc = {};
    for (int k0 = 0; k0 < K_; k0 += 4) {
      const int ka = k0 + 2 * half;
      v2f av, bv;
      av.x = att[i0 + l][ka];
      av.y = att[i0 + l][ka + 1];
      bv.x = xs[ka][c0 + l];           // B[k][n] = x[k][c0+n]
      bv.y = xs[ka + 1][c0 + l];
      acc = wmma_f32_k4(av, bv, acc);
    }
#pragma unroll
    for (int v = 0; v < 8; ++v)
      op[(size_t)(i0 + v + 8 * half) * C_ + (c0 + l)] = acc[v];
  }
}

// ---------------------------------------------------------------------------
extern "C" void kernel_launch(void* const* d_in, const int* in_sizes, int n_in,
                              void* d_out, int out_size, void* d_ws,
                              size_t ws_size, hipStream_t stream) {
  (void)in_sizes; (void)n_in; (void)out_size; (void)ws_size;

  const float* x  = (const float*)d_in[0];
  const float* W1 = (const float*)d_in[1];
  const float* W2 = (const float*)d_in[2];

  float* out = (float*)d_out;                               // B*M*K*C
  float* adj = out + (size_t)B_ * M_ * K_ * C_;             // B*M*K*K

  // scratch: p (B*CH*M) | h (B*H*M); e reuses p (stream-ordered, p dead).
  float* p    = (float*)d_ws;
  float* hbuf = p + (size_t)B_ * CH_ * M_;
  float* e    = p;

  tile_stats_kernel<<<B_ * M_, 128, 0, stream>>>(x, adj, p);
  mlp1_kernel<<<64, 256, 0, stream>>>(W1, p, hbuf);
  mlp2_kernel<<<128, 256, 0, stream>>>(W2, hbuf, e);
  attn_out_kernel<<<B_ * M_, 128, 0, stream>>>(x, adj, e, out);
}